// Diag_78194174591219
// MI455X (gfx1250) — compile-verified
//
#include <hip/hip_runtime.h>
#include <hip/hip_bf16.h>

// out[n, d] = input[n, d] * W[d]   (N = 200000, D = 512, fp32)
//
// Pure streaming op: 820 MB traffic, 0.125 FLOP/byte -> HBM-bound (~35 us at
// 23.3 TB/s). Strategy: b128 (float4) loads/stores with non-temporal hints
// (dataset >> 192 MB L2, single-touch), W slice held in registers per lane,
// global_prefetch_b8 for the next grid-stride iteration.

typedef float v4f __attribute__((ext_vector_type(4)));

#define D_COLS 512
#define VEC_PER_ROW (D_COLS / 4)   // 128 float4 per row
#define BLOCK_THREADS 256
#define GRID_BLOCKS 4096

__global__ __launch_bounds__(BLOCK_THREADS)
void Diag_78194174591219_kernel(const float* __restrict__ in,
                                const float* __restrict__ w,
                                float* __restrict__ out,
                                int n_rows) {
    const int tid        = blockIdx.x * BLOCK_THREADS + threadIdx.x;
    const int col        = tid & (VEC_PER_ROW - 1);   // which float4 within a row
    const int row0       = tid >> 7;                  // tid / VEC_PER_ROW
    const int row_stride = (GRID_BLOCKS * BLOCK_THREADS) >> 7;

    // Per-lane W slice is loop-invariant: load once into registers.
    const v4f wv = *(const v4f*)(w + col * 4);

    const v4f* __restrict__ inv  = (const v4f*)in;
    v4f*       __restrict__ outv = (v4f*)out;

    const size_t idx_stride = (size_t)row_stride * VEC_PER_ROW;

    for (int r = row0; r < n_rows; r += row_stride) {
        const size_t idx = (size_t)r * VEC_PER_ROW + col;

        // Prefetch next iteration's line (speculative; OOB silently dropped).
        // Lowers to global_prefetch_b8 on gfx1250.
        __builtin_prefetch((const void*)(inv + idx + idx_stride), 0, 0);

        // Single-touch streaming data: non-temporal b128 load/store so the
        // 820 MB stream does not thrash the 192 MB L2.
        v4f x = __builtin_nontemporal_load(inv + idx);
        v4f y = x * wv;
        __builtin_nontemporal_store(y, outv + idx);
    }
}

extern "C" void kernel_launch(void* const* d_in, const int* in_sizes, int n_in,
                              void* d_out, int out_size, void* d_ws, size_t ws_size,
                              hipStream_t stream) {
    const float* in = (const float*)d_in[0];   // [N, 512] fp32
    // d_in[1] = A, unused by the reference forward.
    const float* w  = (const float*)d_in[2];   // [512] fp32
    float* out      = (float*)d_out;           // [N, 512] fp32

    const int n_rows = in_sizes[0] / D_COLS;   // 200000

    Diag_78194174591219_kernel<<<GRID_BLOCKS, BLOCK_THREADS, 0, stream>>>(
        in, w, out, n_rows);
}